// GeoEncoder_3478923509786
// MI455X (gfx1250) — compile-verified
//
#include <hip/hip_runtime.h>

// TensoRF triplane encoder, fused sample + WMMA projection for gfx1250 (wave32).
//
// One wave = 16 points. Lane L (L&15 = point row M, L>>4 = K-half) computes the
// exact 2 vm-channel values the V_WMMA_F32_16X16X4_F32 A-fragment layout needs:
//   lanes 0-15 : VGPR0=K0, VGPR1=K1 ; lanes 16-31 : VGPR0=K2, VGPR1=K3
// so sampling feeds the matrix unit with no LDS and no cross-lane traffic.
// 12 K-chunks (RANK=48) x 2 N-tiles (OUT=32) = 24 wmma per wave.

typedef float v2f __attribute__((ext_vector_type(2)));
typedef float v8f __attribute__((ext_vector_type(8)));

#define RES      512
#define RANK     48
#define OUT_DIM  32
#define PLANE_STRIDE (RES * RES)

struct Bi {
    int   o00, o01, o10, o11;
    float w00, w01, w10, w11;
};

__device__ __forceinline__ Bi bi_setup(float gx, float gy) {
    float fx  = (gx + 1.0f) * 0.5f * (float)(RES - 1);
    float fy  = (gy + 1.0f) * 0.5f * (float)(RES - 1);
    float fx0 = fminf(fmaxf(floorf(fx), 0.0f), (float)(RES - 2));
    float fy0 = fminf(fmaxf(floorf(fy), 0.0f), (float)(RES - 2));
    int   x0  = (int)fx0;
    int   y0  = (int)fy0;
    float wx  = fx - fx0;
    float wy  = fy - fy0;
    Bi r;
    r.o00 = y0 * RES + x0;
    r.o01 = r.o00 + 1;
    r.o10 = r.o00 + RES;
    r.o11 = r.o10 + 1;
    r.w00 = (1.0f - wy) * (1.0f - wx);
    r.w01 = (1.0f - wy) * wx;
    r.w10 = wy * (1.0f - wx);
    r.w11 = wy * wx;
    return r;
}

__device__ __forceinline__ void line_setup(float g, int& i0, float& w) {
    float f  = (g + 1.0f) * 0.5f * (float)(RES - 1);
    float fi = fminf(fmaxf(floorf(f), 0.0f), (float)(RES - 2));
    i0 = (int)fi;
    w  = f - fi;
}

__device__ __forceinline__ float bi_sample(const float* __restrict__ pl, const Bi& s) {
    return s.w00 * pl[s.o00] + s.w01 * pl[s.o01] +
           s.w10 * pl[s.o10] + s.w11 * pl[s.o11];
}

__global__ __launch_bounds__(32) void tensorf_fused_wmma_kernel(
    const float* __restrict__ coords,
    const float* __restrict__ aabb,
    const float* __restrict__ plane_xy,
    const float* __restrict__ plane_xz,
    const float* __restrict__ plane_yz,
    const float* __restrict__ line_z,
    const float* __restrict__ line_y,
    const float* __restrict__ line_x,
    const float* __restrict__ Wm,
    const float* __restrict__ bias,
    float* __restrict__ out,
    int n_points)
{
    const int lane = (int)threadIdx.x;   // 0..31
    const int m    = lane & 15;          // point row within tile / output column
    const int hi   = lane >> 4;          // which K-half this lane supplies
    const int base = (int)blockIdx.x * 16;

    int p  = base + m;
    int pc = p < n_points ? p : (n_points - 1);   // clamp gathers; stores predicated

    // ---- L-infinity contraction (matches _contract_linf) ----
    float c[3];
    #pragma unroll
    for (int i = 0; i < 3; ++i) {
        float lo     = aabb[i];
        float hi_v   = aabb[i + 3];
        float center = (lo + hi_v) * 0.5f;
        float halfw  = fmaxf((hi_v - lo) * 0.5f, 1e-6f);
        c[i] = (coords[pc * 3 + i] - center) / halfw;
    }
    float linf  = fmaxf(fabsf(c[0]), fmaxf(fabsf(c[1]), fabsf(c[2])));
    float safe  = fmaxf(linf, 1.0f);
    float scale = (linf > 1.0f) ? (2.0f - 1.0f / safe) / safe : 1.0f;
    #pragma unroll
    for (int i = 0; i < 3; ++i)
        c[i] = fminf(fmaxf(c[i] * scale, -1.0f), 1.0f);
    const float cx = c[0], cy = c[1], cz = c[2];

    // ---- bilinear / linear interpolation setup (shared across all 48 channels) ----
    const Bi sxy = bi_setup(cx, cy);
    const Bi sxz = bi_setup(cx, cz);
    const Bi syz = bi_setup(cy, cz);
    int   izi, iyi, ixi;
    float wz, wy, wx;
    line_setup(cz, izi, wz);
    line_setup(cy, iyi, wy);
    line_setup(cx, ixi, wx);

    v8f acc0 = {};   // output columns 0..15
    v8f acc1 = {};   // output columns 16..31

    #pragma unroll
    for (int kk = 0; kk < RANK / 4; ++kk) {
        const int r0 = kk * 4 + hi * 2;   // this lane's 2 channels for the A-fragment

        v2f a;
        #pragma unroll
        for (int t = 0; t < 2; ++t) {
            const int r = r0 + t;
            const float* pxy_p = plane_xy + (size_t)r * PLANE_STRIDE;
            const float* pxz_p = plane_xz + (size_t)r * PLANE_STRIDE;
            const float* pyz_p = plane_yz + (size_t)r * PLANE_STRIDE;
            float pxy = bi_sample(pxy_p, sxy);
            float pxz = bi_sample(pxz_p, sxz);
            float pyz = bi_sample(pyz_p, syz);
            const float* lz_p = line_z + (size_t)r * RES;
            const float* ly_p = line_y + (size_t)r * RES;
            const float* lx_p = line_x + (size_t)r * RES;
            float lz = lz_p[izi] * (1.0f - wz) + lz_p[izi + 1] * wz;
            float ly = ly_p[iyi] * (1.0f - wy) + ly_p[iyi + 1] * wy;
            float lx = lx_p[ixi] * (1.0f - wx) + lx_p[ixi + 1] * wx;
            a[t] = pxy * lz + pxz * ly + pyz * lx;
        }

        // B-fragment: B[k][n] = W[n][k]; lane supplies n = ntile*16 + m, k = r0, r0+1.
        // W row stride = 48 floats, r0 even -> 8-byte aligned v2f loads.
        v2f b0 = *(const v2f*)(Wm + (size_t)m * RANK + r0);
        v2f b1 = *(const v2f*)(Wm + (size_t)(16 + m) * RANK + r0);

        acc0 = __builtin_amdgcn_wmma_f32_16x16x4_f32(
            false, a, false, b0, (short)0, acc0, false, false);
        acc1 = __builtin_amdgcn_wmma_f32_16x16x4_f32(
            false, a, false, b1, (short)0, acc1, false, false);
    }

    // ---- bias + store. D layout: lanes 0-15 col=m rows 0..7; lanes 16-31 rows 8..15.
    const float bias0 = bias[m];
    const float bias1 = bias[16 + m];
    #pragma unroll
    for (int j = 0; j < 8; ++j) {
        const int row = base + j + 8 * hi;
        if (row < n_points) {
            out[(size_t)row * OUT_DIM + m]      = acc0[j] + bias0;
            out[(size_t)row * OUT_DIM + 16 + m] = acc1[j] + bias1;
        }
    }
}

extern "C" void kernel_launch(void* const* d_in, const int* in_sizes, int n_in,
                              void* d_out, int out_size, void* d_ws, size_t ws_size,
                              hipStream_t stream) {
    (void)n_in; (void)out_size; (void)d_ws; (void)ws_size;

    const float* coords   = (const float*)d_in[0];
    const float* aabb     = (const float*)d_in[1];
    const float* plane_xy = (const float*)d_in[2];
    const float* plane_xz = (const float*)d_in[3];
    const float* plane_yz = (const float*)d_in[4];
    const float* line_z   = (const float*)d_in[5];
    const float* line_y   = (const float*)d_in[6];
    const float* line_x   = (const float*)d_in[7];
    const float* Wm       = (const float*)d_in[8];
    const float* bias     = (const float*)d_in[9];
    float* out = (float*)d_out;

    const int n     = in_sizes[0] / 3;      // N points
    const int tiles = (n + 15) / 16;        // one wave (32 threads) per 16 points

    tensorf_fused_wmma_kernel<<<tiles, 32, 0, stream>>>(
        coords, aabb, plane_xy, plane_xz, plane_yz,
        line_z, line_y, line_x, Wm, bias, out, n);
}